// EdgeConv_69956427317653
// MI455X (gfx1250) — compile-verified
//
#include <hip/hip_runtime.h>
#include <stdint.h>

#define BATCH 2
#define CH    64
#define NPT   8192
#define KNN_K 20
#define COUT  64
#define CIN2  128
#define NEGSL 0.2f
#define BNEPS 1e-5f

typedef float v2f __attribute__((ext_vector_type(2)));
typedef float v4f __attribute__((ext_vector_type(4)));
typedef float v8f __attribute__((ext_vector_type(8)));

// CDNA5 fp32 WMMA: D(16x16) = A(16x4) * B(4x16) + C
__device__ __forceinline__ v8f wmma_f32(v2f a, v2f b, v8f c) {
  return __builtin_amdgcn_wmma_f32_16x16x4_f32(false, a, false, b, (short)0, c, false, false);
}

// order-preserving float <-> uint encoding (for exact LDS integer max/min atomics)
__device__ __forceinline__ unsigned fenc(float f) {
  unsigned u = __float_as_uint(f);
  return (u & 0x80000000u) ? ~u : (u | 0x80000000u);
}
__device__ __forceinline__ float fdec(unsigned u) {
  return __uint_as_float((u & 0x80000000u) ? (u ^ 0x80000000u) : ~u);
}

// ---------------- Kernel A: transpose x -> xt[b][n][c], squared norms ----------
__global__ void prep_kernel(const float* __restrict__ x,
                            float* __restrict__ xt, float* __restrict__ sq) {
  int gp = blockIdx.x * blockDim.x + threadIdx.x;   // [0, B*N)
  int b = gp >> 13, n = gp & (NPT - 1);
  const float* xb = x + (size_t)b * CH * NPT + n;
  float acc = 0.f;
  #pragma unroll 8
  for (int c = 0; c < CH; ++c) {
    float v = xb[(size_t)c * NPT];
    acc += v * v;
    xt[(size_t)gp * CH + c] = v;
  }
  sq[gp] = acc;
}

// ---------------- Kernel B: distances via f32 WMMA + streaming top-K ------------
// One wave (32 threads) per 16-row block. Lane L: row = L&15, column-half = L>>4.
// B operands double-buffered in registers: loads for tile ct+1 overlap the
// LDS-store/barrier/top-K-scan phase of tile ct.
__launch_bounds__(32)
__global__ void knn_kernel(const float* __restrict__ xt, const float* __restrict__ sq,
                           int* __restrict__ knn) {
  __shared__ float tile[16 * 16];        // distance tile, col-major [col][row]
  __shared__ float mv[32 * KNN_K];       // merge buffers
  __shared__ int   mi[32 * KNN_K];

  const int L  = threadIdx.x;
  const int rr = L & 15, hh = L >> 4;
  const int b  = blockIdx.x >> 9;
  const int r0 = (blockIdx.x & 511) << 4;
  const int bofs = b << 13;

  // Preload A operands for all 16 K-steps (row block fixed): lane holds
  // A[M=rr][k0+2*hh .. +1] per ISA 16x4 f32 A layout.
  v2f a[16];
  const float* arow = xt + (size_t)(bofs + r0 + rr) * CH + 2 * hh;
  #pragma unroll
  for (int kk = 0; kk < 16; ++kk) a[kk] = *(const v2f*)(arow + kk * 4);
  const float sqn = sq[bofs + r0 + rr];

  float vals[KNN_K]; int idxs[KNN_K];
  #pragma unroll
  for (int q = 0; q < KNN_K; ++q) { vals[q] = -__builtin_inff(); idxs[q] = 0; }

  // B operand: B[k][n]=xt[c0+n][k] -> identical per-lane pattern as A (Gram matrix)
  const float* xbase = xt + (size_t)bofs * CH + 2 * hh;

  auto loadB = [&](int ct, v2f (&bb)[16]) {
    const float* brow = xbase + (size_t)(((ct << 4) + rr)) * CH;
    #pragma unroll
    for (int kk = 0; kk < 16; ++kk) bb[kk] = *(const v2f*)(brow + kk * 4);
  };

  auto step = [&](int ct, v2f (&cur)[16], v2f (&nxt)[16]) {
    const int c0 = ct << 4;
    // sq values for this tile's scan (issued early, overlap the WMMA chain)
    v4f sq0 = *(const v4f*)(sq + bofs + c0 + hh * 8);
    v4f sq1 = *(const v4f*)(sq + bofs + c0 + hh * 8 + 4);
    float sqt[8];
    #pragma unroll
    for (int j = 0; j < 4; ++j) { sqt[j] = sq0[j]; sqt[j + 4] = sq1[j]; }

    v8f acc = {};
    #pragma unroll
    for (int kk = 0; kk < 16; ++kk) acc = wmma_f32(a[kk], cur[kk], acc);

    // prefetch next tile's B operands; latency hidden by store+scan below
    loadB((ct + 1) & 511, nxt);

    // Store accumulator to LDS col-major: lane holds col N=rr, rows 8*hh..8*hh+7
    v4f lo = __builtin_shufflevector(acc, acc, 0, 1, 2, 3);
    v4f hi = __builtin_shufflevector(acc, acc, 4, 5, 6, 7);
    float* p = &tile[rr * 16 + hh * 8];
    *(v4f*)p       = lo;
    *(v4f*)(p + 4) = hi;
    __syncthreads();

    // Lane scans row rr over its 8 columns of this tile
    #pragma unroll
    for (int j = 0; j < 8; ++j) {
      int ci = hh * 8 + j;
      int colg = c0 + ci;
      float f = tile[ci * 16 + rr];
      float s = 2.f * f - sqt[j] - sqn;            // neg_dist
      if (s > vals[0]) {                            // rare branch
        float old19 = vals[KNN_K - 1];
        #pragma unroll
        for (int t = 0; t < KNN_K - 1; ++t) {
          bool sh = s > vals[t + 1];
          float kv = vals[t]; int ki = idxs[t];
          vals[t] = sh ? vals[t + 1] : fmaxf(s, kv);
          idxs[t] = sh ? idxs[t + 1] : ((kv >= s) ? ki : colg);
        }
        if (s > old19) { vals[KNN_K - 1] = s; idxs[KNN_K - 1] = colg; }
      }
    }
    __syncthreads();
  };

  v2f bufA[16], bufB[16];
  loadB(0, bufA);
  for (int ct = 0; ct < NPT / 16; ct += 2) {
    step(ct,     bufA, bufB);
    step(ct + 1, bufB, bufA);
  }

  // Merge the two lanes (L, L+16) that scanned the same row
  #pragma unroll
  for (int q = 0; q < KNN_K; ++q) { mv[L * KNN_K + q] = vals[q]; mi[L * KNN_K + q] = idxs[q]; }
  __syncthreads();
  if (L < 16) {
    const float* va = &mv[L * KNN_K];          const int* ia = &mi[L * KNN_K];
    const float* vb = &mv[(L + 16) * KNN_K];   const int* ib = &mi[(L + 16) * KNN_K];
    int pa = KNN_K - 1, pb = KNN_K - 1;
    int* outp = knn + (size_t)(bofs + r0 + L) * KNN_K;
    for (int q = 0; q < KNN_K; ++q) {
      int sa = pa < 0 ? 0 : pa, sb = pb < 0 ? 0 : pb;
      float fa = (pa >= 0) ? va[sa] : -__builtin_inff();
      float fb = (pb >= 0) ? vb[sb] : -__builtin_inff();
      bool ta = fa >= fb;
      outp[q] = ta ? ia[sa] : ib[sb];
      if (ta) --pa; else --pb;
    }
  }
}

// ---------------- Kernel C: edge conv (f32 WMMA) + leaky + stats + k-max/min ----
// 4 waves/WG; each wave owns 4 points (80 edge rows = 5 tiles of 16).
__launch_bounds__(128)
__global__ void conv_kernel(const float* __restrict__ xt, const int* __restrict__ knn,
                            const float* __restrict__ W, const float* __restrict__ bias,
                            float* __restrict__ mxp, float* __restrict__ mnp,
                            float* __restrict__ part) {
  __shared__ float    s_edge[4][16 * CIN2];     // 32 KB: per-wave edge tile
  __shared__ unsigned s_umax[4][4][COUT];       // per-wave, per-point, per-channel
  __shared__ unsigned s_umin[4][4][COUT];
  __shared__ float    s_psum[4][16][8];         // [ot][rr][w*2+hh]
  __shared__ float    s_psq [4][16][8];

  const int tid = threadIdx.x;
  const int w = tid >> 5, L = tid & 31;
  const int rr = L & 15, hh = L >> 4;
  const int wfp = blockIdx.x * 16 + w * 4;      // wave's first global point

  {
    unsigned* um = &s_umax[0][0][0];
    unsigned* un = &s_umin[0][0][0];
    for (int i = tid; i < 4 * 4 * COUT; i += 128) { um[i] = 0u; un[i] = 0xFFFFFFFFu; }
  }
  __syncthreads();

  float fsum[4] = {0.f, 0.f, 0.f, 0.f}, fsq[4] = {0.f, 0.f, 0.f, 0.f};
  float* edge = &s_edge[w][0];

  #pragma unroll 1
  for (int t = 0; t < 5; ++t) {
    // ---- stage 16 edge rows: [diff(64) | center(64)] ----
    {
      int j = L >> 1, half = L & 1;
      int le = t * 16 + j;
      int p = le / 20, k = le - 20 * p;
      int gp = wfp + p;
      int bb = gp >> 13;
      int m = knn[(size_t)gp * KNN_K + k];
      const float* xn = xt + (size_t)gp * CH;
      const float* xm = xt + (size_t)((bb << 13) + m) * CH;
      float* er = edge + j * CIN2;
      if (half == 0) {
        #pragma unroll
        for (int c = 0; c < CH; c += 4)
          *(v4f*)(er + c) = *(const v4f*)(xm + c) - *(const v4f*)(xn + c);
      } else {
        #pragma unroll
        for (int c = 0; c < CH; c += 4)
          *(v4f*)(er + CH + c) = *(const v4f*)(xn + c);
      }
    }
    __syncthreads();

    // group boundary inside this lane's 8 rows (rows base..base+7, points change at /20)
    int le0 = t * 16 + hh * 8;
    int pA = le0 / 20;
    int vth = (pA + 1) * 20 - le0;                 // rows v >= vth belong to point pA+1
    bool hasB = vth < 8;

    #pragma unroll 1
    for (int ot = 0; ot < 4; ++ot) {
      const int o = ot * 16 + rr;                  // lane's output channel
      const float* wrow = W + (size_t)o * CIN2 + 2 * hh;
      const float* arow = edge + rr * CIN2 + 2 * hh;
      v8f acc = {};
      // grouped loads (clause-able) then back-to-back WMMAs: amortize waits 8x
      #pragma unroll
      for (int g = 0; g < 4; ++g) {
        v2f av[8], bv[8];
        #pragma unroll
        for (int u = 0; u < 8; ++u) {
          av[u] = *(const v2f*)(arow + (g * 8 + u) * 4);
          bv[u] = *(const v2f*)(wrow + (g * 8 + u) * 4);
        }
        #pragma unroll
        for (int u = 0; u < 8; ++u) acc = wmma_f32(av[u], bv[u], acc);
      }
      float bo = bias[o];
      float gAmax = -__builtin_inff(), gAmin = __builtin_inff();
      float gBmax = -__builtin_inff(), gBmin = __builtin_inff();
      float lsum = 0.f, lsq = 0.f;
      #pragma unroll
      for (int v = 0; v < 8; ++v) {
        float h = acc[v] + bo;
        h = (h >= 0.f) ? h : NEGSL * h;            // LeakyReLU
        lsum += h; lsq += h * h;
        bool inB = (v >= vth);
        gAmax = fmaxf(gAmax, inB ? -__builtin_inff() : h);
        gAmin = fminf(gAmin, inB ?  __builtin_inff() : h);
        gBmax = fmaxf(gBmax, inB ? h : -__builtin_inff());
        gBmin = fminf(gBmin, inB ? h :  __builtin_inff());
      }
      fsum[ot] += lsum; fsq[ot] += lsq;
      atomicMax(&s_umax[w][pA][o], fenc(gAmax));   // ds_max_u32 (exact, order-free)
      atomicMin(&s_umin[w][pA][o], fenc(gAmin));
      if (hasB) {
        atomicMax(&s_umax[w][pA + 1][o], fenc(gBmax));
        atomicMin(&s_umin[w][pA + 1][o], fenc(gBmin));
      }
    }
    __syncthreads();
  }

  // ---- deterministic per-WG stats partials (fixed-order reduction) ----
  #pragma unroll
  for (int ot = 0; ot < 4; ++ot) {
    s_psum[ot][rr][w * 2 + hh] = fsum[ot];
    s_psq [ot][rr][w * 2 + hh] = fsq[ot];
  }
  __syncthreads();
  if (tid < COUT) {
    float s = 0.f, s2 = 0.f;
    #pragma unroll
    for (int q = 0; q < 8; ++q) { s += s_psum[tid >> 4][tid & 15][q]; s2 += s_psq[tid >> 4][tid & 15][q]; }
    part[(size_t)blockIdx.x * 128 + tid]      = s;
    part[(size_t)blockIdx.x * 128 + 64 + tid] = s2;
  }

  // ---- write per-(b,o,n) max/min planes (exclusive ownership, no global atomics) ----
  #pragma unroll
  for (int it = 0; it < 8; ++it) {
    int q = L + 32 * it;
    int p = q >> 6, o = q & 63;
    int gp = wfp + p;
    int bb = gp >> 13, n = gp & (NPT - 1);
    size_t oi = ((size_t)(bb * COUT + o) << 13) + n;
    mxp[oi] = fdec(s_umax[w][p][o]);
    mnp[oi] = fdec(s_umin[w][p][o]);
  }
}

// ---------------- Kernel S: fixed-order final stats reduction -------------------
__global__ void stats_kernel(const float* __restrict__ part, float* __restrict__ gst) {
  int t = threadIdx.x;            // 128 threads: [0,64)=sum, [64,128)=sumsq
  float s = 0.f;
  for (int g = 0; g < 1024; ++g) s += part[(size_t)g * 128 + t];
  gst[t] = s;
}

// ---------------- Kernel D: BN finalize + max-over-k via monotone trick ---------
__global__ void out_kernel(const float* __restrict__ mxp, const float* __restrict__ mnp,
                           const float* __restrict__ gst,
                           const float* __restrict__ gamma, const float* __restrict__ beta,
                           float* __restrict__ outp) {
  int i = blockIdx.x * blockDim.x + threadIdx.x;   // [0, B*COUT*N), layout (b,o,n)
  int o = (i >> 13) & 63;
  const float cnt = (float)(BATCH * NPT * KNN_K);
  float mean = gst[o] / cnt;
  float var  = gst[COUT + o] / cnt - mean * mean;
  float scale = gamma[o] * rsqrtf(var + BNEPS);
  float sel = (scale >= 0.f) ? mxp[i] : mnp[i];    // max of affine = affine of max/min
  outp[i] = (sel - mean) * scale + beta[o];
}

extern "C" void kernel_launch(void* const* d_in, const int* in_sizes, int n_in,
                              void* d_out, int out_size, void* d_ws, size_t ws_size,
                              hipStream_t stream) {
  const float* x     = (const float*)d_in[0];
  const float* W     = (const float*)d_in[1];
  const float* bias  = (const float*)d_in[2];
  const float* gamma = (const float*)d_in[3];
  const float* beta  = (const float*)d_in[4];
  float* outp = (float*)d_out;

  // workspace layout (floats / ints), ~14.5 MB total
  float* xt  = (float*)d_ws;                         // B*N*C
  float* sq  = xt + (size_t)BATCH * NPT * CH;        // B*N
  int*   knn = (int*)(sq + (size_t)BATCH * NPT);     // B*N*K
  float* mxp = (float*)(knn + (size_t)BATCH * NPT * KNN_K);  // B*COUT*N
  float* mnp = mxp + (size_t)BATCH * COUT * NPT;     // B*COUT*N
  float* gst = mnp + (size_t)BATCH * COUT * NPT;     // 128
  float* part = gst + 128;                           // 1024*128

  prep_kernel <<<(BATCH * NPT) / 256, 256, 0, stream>>>(x, xt, sq);
  knn_kernel  <<<BATCH * (NPT / 16), 32, 0, stream>>>(xt, sq, knn);
  conv_kernel <<<(BATCH * NPT) / 16, 128, 0, stream>>>(xt, knn, W, bias, mxp, mnp, part);
  stats_kernel<<<1, 128, 0, stream>>>(part, gst);
  out_kernel  <<<(BATCH * COUT * NPT) / 256, 256, 0, stream>>>(mxp, mnp, gst, gamma, beta, outp);
}